// InterLKD_2448131359327
// MI455X (gfx1250) — compile-verified
//
#include <hip/hip_runtime.h>
#include <math.h>

// ---------------- static problem configuration ----------------
#define BB     8
#define CC     32
#define HH     512
#define WW     512
#define HW     (HH * WW)        // 262144 = 2^18
#define DIM    2048             // 32*8*8
#define NREG   24576
#define QS     4096
#define CSIZE  256
#define KNEG   768              // 3 * 256
#define ROWS   32               // region rows per block
#define KC     1024             // K-chunk staged in LDS
#define KPC    1026             // padded chunk row stride (floats), keeps b64 loads 8B aligned
#define LDSF   (2 * ROWS * KPC) // total LDS floats (As + At) = 65664 -> 262656 B

typedef __attribute__((ext_vector_type(2))) float v2f;
typedef __attribute__((ext_vector_type(4))) float v4f;
typedef __attribute__((ext_vector_type(8))) float v8f;

// ---------------- kernel 1: inverse channel-L2-norm maps ----------------
__global__ __launch_bounds__(256)
void norm_kernel(const float* __restrict__ s, const float* __restrict__ t,
                 float* __restrict__ inv_s, float* __restrict__ inv_t) {
  int p = blockIdx.x * 256 + threadIdx.x;        // 0 .. B*H*W-1
  int b = p >> 18;
  int off = p & (HW - 1);
  const float* ps = s + (size_t)b * CC * HW + off;
  const float* pt = t + (size_t)b * CC * HW + off;
  float as = 0.f, at = 0.f;
#pragma unroll 8
  for (int c = 0; c < CC; ++c) {
    float vs = ps[(size_t)c * HW];
    float vt = pt[(size_t)c * HW];
    as = fmaf(vs, vs, as);
    at = fmaf(vt, vt, at);
  }
  inv_s[p] = 1.0f / fmaxf(sqrtf(as), 1e-12f);
  inv_t[p] = 1.0f / fmaxf(sqrtf(at), 1e-12f);
}

// ---------------- kernel 2: curriculum rank selection (stable argsort slice) ----------------
__global__ __launch_bounds__(256)
void select_kernel(const float* __restrict__ qs, const int* __restrict__ epoch_p,
                   int* __restrict__ sel) {
  const int cls = blockIdx.x;                    // 0..2
  const int tid = threadIdx.x;
  const int e = epoch_p[0];
  // s1 = int(2048*0.6)=1228 ; step1 = ((2048-1-1228+1)-256)/500
  // s2 = int(4096*0.6)=2457 ; step2 = ((4096-1-2457+1)-256)/500
  const double step1 = (820.0 - 256.0) / 500.0;
  const double step2 = (1639.0 - 256.0) / 500.0;
  const int s1 = 1228 + (int)(step1 * (double)e);
  const int s2 = 2457 + (int)(step2 * (double)e);
  const int n   = (cls == 2) ? QS : 2048;
  const int beg = (cls == 2) ? s2 : s1;
  const float* v = qs + (size_t)cls * QS;
  for (int i = tid; i < n; i += 256) {
    float vi = v[i];
    int rank = 0;
    for (int j = 0; j < n; ++j) {
      float vj = v[j];
      rank += (vj < vi) || (vj == vi && j < i);  // stable ascending rank
    }
    int r = rank - beg;
    if (r >= 0 && r < CSIZE) sel[cls * CSIZE + r] = i;
  }
}

// ---------------- kernel 3: gather negatives into b128-fragment-packed layout ----------------
// Layout: P[q][j][h][4], q = k/8, h = lane-half (khalf/2).
//   P[q][j][h] = { Xc[j][8q+2h], Xc[j][8q+2h+1], Xc[j][8q+4+2h], Xc[j][8q+4+2h+1] } * 10
// One aligned b128 load per lane yields B fragments for TWO consecutive k-quads.
__global__ __launch_bounds__(256)
void gather_pack_kernel(const float* __restrict__ rq, const int* __restrict__ sel,
                        float* __restrict__ bpack) {
  const int j = blockIdx.x;                      // 0..767
  const int cls = j >> 8;
  const int idx = sel[j];
  const float* src = rq + ((size_t)cls * QS + (size_t)idx) * DIM;
  for (int k = threadIdx.x; k < DIM; k += 256) {
    int q   = k >> 3;
    int r   = k & 7;
    int sub = r >> 2;          // which k-quad of the pair
    int h   = (r >> 1) & 1;    // lane half
    int b   = r & 1;
    size_t dst = (((size_t)q * KNEG + j) * 2 + h) * 4 + (2 * sub + b);
    bpack[dst] = src[k] * 10.0f;                 // fold 1/T_C = 10
  }
}

// ---------------- kernel 4: fused gather + dual fp32-WMMA GEMM + softmax-KL ----------------
__device__ __forceinline__
void stage_chunk(const float* __restrict__ feats, const float* __restrict__ inv,
                 float* __restrict__ Astage, int rb, int kbase,
                 const int* __restrict__ bb, const int* __restrict__ by,
                 const int* __restrict__ bx, int tid) {
  for (int epos = tid; epos < ROWS * KC; epos += 256) {
    int mm = epos >> 10;                 // region row 0..31
    int k  = kbase + (epos & (KC - 1));  // global k 0..2047
    int r = rb + mm;
    int b = bb[r], y0 = by[r], x0 = bx[r];
    int c  = k >> 6;
    int yy = (k >> 3) & 7;
    int xx = k & 7;
    int y = y0 + yy, x = x0 + xx;
    size_t fp = (((size_t)(b * CC + c)) * HH + y) * WW + x;
    size_t ip = ((size_t)b * HH + y) * WW + x;
    Astage[mm * KPC + (epos & (KC - 1))] = feats[fp] * inv[ip];
  }
}

__device__ __forceinline__
void gemm_chunk(const float* __restrict__ As, const float* __restrict__ At,
                const float* __restrict__ bq_base,
                int m, int khalf, v8f (&accS)[2][6], v8f (&accT)[2][6]) {
  for (int kb = 0; kb < KC; kb += 8) {
    // A fragments: 2 tensors x 2 row-tiles x 2 k-quads (pairs merge to ds_load_2addr_b64)
    v2f as00 = *(const v2f*)(As + m * KPC + kb + khalf);
    v2f as01 = *(const v2f*)(As + m * KPC + kb + 4 + khalf);
    v2f as10 = *(const v2f*)(As + (m + 16) * KPC + kb + khalf);
    v2f as11 = *(const v2f*)(As + (m + 16) * KPC + kb + 4 + khalf);
    v2f at00 = *(const v2f*)(At + m * KPC + kb + khalf);
    v2f at01 = *(const v2f*)(At + m * KPC + kb + 4 + khalf);
    v2f at10 = *(const v2f*)(At + (m + 16) * KPC + kb + khalf);
    v2f at11 = *(const v2f*)(At + (m + 16) * KPC + kb + 4 + khalf);
    const float* xq = bq_base + (size_t)(kb >> 3) * (KNEG * 8);
#pragma unroll
    for (int t6 = 0; t6 < 6; ++t6) {
      v4f bq = *(const v4f*)(xq + t6 * 128);     // ONE b128: B frags for both k-quads
      v2f b0; b0.x = bq.x; b0.y = bq.y;
      v2f b1; b1.x = bq.z; b1.y = bq.w;
      accS[0][t6] = __builtin_amdgcn_wmma_f32_16x16x4_f32(false, as00, false, b0, (short)0, accS[0][t6], false, false);
      accS[0][t6] = __builtin_amdgcn_wmma_f32_16x16x4_f32(false, as01, false, b1, (short)0, accS[0][t6], false, false);
      accS[1][t6] = __builtin_amdgcn_wmma_f32_16x16x4_f32(false, as10, false, b0, (short)0, accS[1][t6], false, false);
      accS[1][t6] = __builtin_amdgcn_wmma_f32_16x16x4_f32(false, as11, false, b1, (short)0, accS[1][t6], false, false);
      accT[0][t6] = __builtin_amdgcn_wmma_f32_16x16x4_f32(false, at00, false, b0, (short)0, accT[0][t6], false, false);
      accT[0][t6] = __builtin_amdgcn_wmma_f32_16x16x4_f32(false, at01, false, b1, (short)0, accT[0][t6], false, false);
      accT[1][t6] = __builtin_amdgcn_wmma_f32_16x16x4_f32(false, at10, false, b0, (short)0, accT[1][t6], false, false);
      accT[1][t6] = __builtin_amdgcn_wmma_f32_16x16x4_f32(false, at11, false, b1, (short)0, accT[1][t6], false, false);
    }
  }
}

__global__ __launch_bounds__(256)
void gemm_kl_kernel(const float* __restrict__ s_feats, const float* __restrict__ t_feats,
                    const float* __restrict__ inv_s, const float* __restrict__ inv_t,
                    const float* __restrict__ bpack,
                    const int* __restrict__ bb, const int* __restrict__ by,
                    const int* __restrict__ bx,
                    float* __restrict__ partials) {
  extern __shared__ float lds[];         // 262656 B (within 320KB WGP LDS)
  const int tid  = threadIdx.x;
  const int wave = tid >> 5;             // 8 waves
  const int lane = tid & 31;             // wave32
  const int rb = blockIdx.x * ROWS;      // 32 region rows per block
  const int cb = wave * 96;              // 96 negatives (6 WMMA tiles) per wave
  const int m = lane & 15;
  const int khalf = (lane >> 4) << 1;    // lanes 0-15: K+{0,1}; lanes 16-31: K+{2,3}

  float* As = lds;
  float* At = lds + ROWS * KPC;

  v8f accS[2][6], accT[2][6];
  v8f zero = {};
#pragma unroll
  for (int rt = 0; rt < 2; ++rt)
#pragma unroll
    for (int i = 0; i < 6; ++i) { accS[rt][i] = zero; accT[rt][i] = zero; }

  // per-lane packed-B base: column (cb+col), this lane-half's pair slot
  const float* bq_base = bpack + ((size_t)(cb + m) * 2 + (khalf >> 1)) * 4;

  // two K-chunks; both tensors staged together so each B fragment feeds 2 tensors
  for (int kbase = 0; kbase < DIM; kbase += KC) {
    stage_chunk(s_feats, inv_s, As, rb, kbase, bb, by, bx, tid);
    stage_chunk(t_feats, inv_t, At, rb, kbase, bb, by, bx, tid);
    __syncthreads();
    gemm_chunk(As, At, bq_base + (size_t)(kbase >> 3) * (KNEG * 8), m, khalf, accS, accT);
    __syncthreads();
  }

  // epilogue: overlay stage buffer with logits [32][768] x2 (196608 B <= 262656 B)
  float* slog = lds;
  float* tlog = lds + ROWS * KNEG;
  {
    const int Mbase = (lane >> 4) << 3;  // D layout: lanes 16-31 hold M+8
    const int N = lane & 15;
#pragma unroll
    for (int rt = 0; rt < 2; ++rt) {
#pragma unroll
      for (int t6 = 0; t6 < 6; ++t6) {
        int colw = cb + t6 * 16 + N;
#pragma unroll
        for (int v = 0; v < 8; ++v) {
          slog[(rt * 16 + Mbase + v) * KNEG + colw] = accS[rt][t6][v];
          tlog[(rt * 16 + Mbase + v) * KNEG + colw] = accT[rt][t6][v];
        }
      }
    }
  }
  __syncthreads();

  // softmax-KL per row: KL(p_t || p_s), 4 rows per wave, butterfly reductions
  float waveSum = 0.0f;
  for (int rr = 0; rr < 4; ++rr) {
    int row = wave * 4 + rr;
    const float* sr = slog + row * KNEG;
    const float* tr = tlog + row * KNEG;
    float mS = -3.4e38f, mT = -3.4e38f;
    for (int j = lane; j < KNEG; j += 32) {
      mS = fmaxf(mS, sr[j]);
      mT = fmaxf(mT, tr[j]);
    }
#pragma unroll
    for (int o = 16; o > 0; o >>= 1) {
      mS = fmaxf(mS, __shfl_xor(mS, o, 32));
      mT = fmaxf(mT, __shfl_xor(mT, o, 32));
    }
    float eS = 0.f, eT = 0.f;
    for (int j = lane; j < KNEG; j += 32) {
      eS += __expf(sr[j] - mS);
      eT += __expf(tr[j] - mT);
    }
#pragma unroll
    for (int o = 16; o > 0; o >>= 1) {
      eS += __shfl_xor(eS, o, 32);
      eT += __shfl_xor(eT, o, 32);
    }
    float lseS = __logf(eS) + mS;
    float lseT = __logf(eT) + mT;
    float a = 0.f;
    for (int j = lane; j < KNEG; j += 32) {
      float lt = tr[j] - lseT;
      float ls = sr[j] - lseS;
      a += __expf(lt) * (lt - ls);
    }
#pragma unroll
    for (int o = 16; o > 0; o >>= 1) a += __shfl_xor(a, o, 32);
    waveSum += a;
  }
  __syncthreads();
  if (lane == 0) lds[2 * ROWS * KNEG + wave] = waveSum;  // scratch past logits (< LDSF)
  __syncthreads();
  if (tid == 0) {
    float x = 0.f;
    for (int w = 0; w < 8; ++w) x += lds[2 * ROWS * KNEG + w];
    partials[blockIdx.x] = x;
  }
}

// ---------------- kernel 5: deterministic final reduction ----------------
__global__ __launch_bounds__(256)
void final_reduce_kernel(const float* __restrict__ partials, float* __restrict__ out) {
  __shared__ float sh[256];
  float a = 0.f;
  for (int i = threadIdx.x; i < NREG / ROWS; i += 256) a += partials[i];
  sh[threadIdx.x] = a;
  __syncthreads();
  for (int s = 128; s > 0; s >>= 1) {
    if (threadIdx.x < s) sh[threadIdx.x] += sh[threadIdx.x + s];
    __syncthreads();
  }
  if (threadIdx.x == 0) out[0] = sh[0] / (float)NREG;   // batchmean, T_KD=1
}

// ---------------- launcher ----------------
extern "C" void kernel_launch(void* const* d_in, const int* in_sizes, int n_in,
                              void* d_out, int out_size, void* d_ws, size_t ws_size,
                              hipStream_t stream) {
  (void)in_sizes; (void)n_in; (void)out_size; (void)ws_size;
  const float* s_feats = (const float*)d_in[0];
  const float* t_feats = (const float*)d_in[1];
  const float* rqueue  = (const float*)d_in[2];
  const float* qscores = (const float*)d_in[3];
  const int*   box_b   = (const int*)d_in[4];
  const int*   box_y   = (const int*)d_in[5];
  const int*   box_x   = (const int*)d_in[6];
  const int*   epoch   = (const int*)d_in[7];
  float* out = (float*)d_out;

  // workspace layout (~23 MB total)
  char* ws = (char*)d_ws;
  float* inv_s    = (float*)(ws + 0);                       // 8 MB
  float* inv_t    = (float*)(ws + 8388608);                 // 8 MB
  float* bpack    = (float*)(ws + 16777216);                // 6 MB  [DIM/8][KNEG][2][4]
  int*   sel      = (int*)  (ws + 23068672);                // 3 KB
  float* partials = (float*)(ws + 23071744);                // 3 KB

  norm_kernel<<<(BB * HW) / 256, 256, 0, stream>>>(s_feats, t_feats, inv_s, inv_t);
  select_kernel<<<3, 256, 0, stream>>>(qscores, epoch, sel);
  gather_pack_kernel<<<KNEG, 256, 0, stream>>>(rqueue, sel, bpack);
  gemm_kl_kernel<<<NREG / ROWS, 256, LDSF * sizeof(float), stream>>>(
      s_feats, t_feats, inv_s, inv_t, bpack, box_b, box_y, box_x, partials);
  final_reduce_kernel<<<1, 256, 0, stream>>>(partials, out);
}